// GenericHeteroGNN_1125281432267
// MI455X (gfx1250) — compile-verified
//
#include <hip/hip_runtime.h>

#define NR   100000   // robot nodes
#define NB   100000   // ball nodes
#define NE   600000   // edges per edge type
#define DIN  128
#define DH   128
#define DOUT 64

typedef __attribute__((ext_vector_type(2))) float v2f;
typedef __attribute__((ext_vector_type(8))) float v8f;

// ---------------------------------------------------------------------------
// Edge scatter: agg[dst,:] += x[src,:], deg[dst] += 1 (optional).
// One wave32 per edge; each lane handles 4 consecutive features (one 512 B
// coalesced float4 gather per edge, 4x global_atomic_add_f32 scatter). The
// 51 MB accumulators are L2-resident (192 MB L2) so atomics resolve in L2.
// ---------------------------------------------------------------------------
__global__ __launch_bounds__(256)
void scatter_accum(const float* __restrict__ xsrc,
                   const int* __restrict__ src, const int* __restrict__ dst,
                   float* __restrict__ agg, float* __restrict__ deg, int nE)
{
    int t    = blockIdx.x * blockDim.x + threadIdx.x;
    int e    = t >> 5;
    int lane = t & 31;
    if (e >= nE) return;
    int s = src[e];
    int d = dst[e];
    const float4 v = *(const float4*)(xsrc + (size_t)s * DIN + lane * 4);
    float* o = agg + (size_t)d * DIN + lane * 4;
    atomicAdd(o + 0, v.x);
    atomicAdd(o + 1, v.y);
    atomicAdd(o + 2, v.z);
    atomicAdd(o + 3, v.w);
    if (deg && lane == 0) atomicAdd(deg + d, 1.0f);
}

__global__ __launch_bounds__(256)
void invdeg_kernel(float* __restrict__ deg, int n)
{
    int i = blockIdx.x * blockDim.x + threadIdx.x;
    if (i < n) deg[i] = 1.0f / fmaxf(deg[i], 1.0f);
}

// ---------------------------------------------------------------------------
// Fused HeteroConv GEMM, LDS-staged:
//   out = act( s0*A0@W0 + A1@W1 [+ s2*A2@W2 + A3@W3] + bias0 [+ bias1] )
//
// Block = 128 threads = 4 waves, one 16-row M tile, 64 output columns
// (wave w owns columns [blockIdx.y*64 + w*16, +16)). Per term:
//   - A tile (16 x K f32, 8 KB) staged coalesced -> LDS (b128 stores),
//     shared by all 4 waves (4x reuse vs. per-wave global fragment loads).
//   - W tile staged TRANSPOSED (WT[col][k], 64 x K, 33 KB) so each lane's
//     B fragment {k+2h, k+2h+1} is one contiguous ds_load_b64.
//   - LDS row stride K+4 = 132 floats: 16 B-aligned rows for b128 staging,
//     and bank-conflict-free b64 fragment reads (bank step 4/row).
// Inner loop: 2x ds_load_b64 + v_pk_mul_f32 (mean scale) + v_wmma per step.
// ---------------------------------------------------------------------------
template<int K>
__global__ __launch_bounds__(128)
void sage_wmma_gemm(const float* __restrict__ A0, const float* __restrict__ W0,
                    const float* __restrict__ s0,
                    const float* __restrict__ A1, const float* __restrict__ W1,
                    const float* __restrict__ A2, const float* __restrict__ W2,
                    const float* __restrict__ s2,
                    const float* __restrict__ A3, const float* __restrict__ W3,
                    const float* __restrict__ bias0, const float* __restrict__ bias1,
                    float* __restrict__ out, int N, int relu)
{
    constexpr int LDA = K + 4;            // padded LDS row stride (floats)
    __shared__ float As[16 * LDA];        // 16-row A tile
    __shared__ float WTs[64 * LDA];       // 64-col transposed W tile

    const int lane = threadIdx.x & 31;
    const int wave = threadIdx.x >> 5;
    const int half = lane >> 4;           // 0: lanes 0-15, 1: lanes 16-31
    const int lm   = lane & 15;
    const int m0   = blockIdx.x * 16;
    const int cb0  = blockIdx.y * 64;     // first column owned by this block
    const int col  = cb0 + wave * 16 + lm;
    const int row  = m0 + lm;

    const float* Aarr[4] = { A0, A1, A2, A3 };
    const float* Warr[4] = { W0, W1, W2, W3 };
    const float* Sarr[4] = { s0, nullptr, s2, nullptr };

    v8f acc = {};

#pragma unroll
    for (int t = 0; t < 4; ++t) {
        const float* A = Aarr[t];
        if (!A) continue;                  // uniform across block
        const float* W = Warr[t];
        const float* scale = Sarr[t];

        __syncthreads();                   // previous term's LDS reads done

        // ---- stage A tile: 16 rows x K, coalesced float4 -> ds_store_b128
        for (int f = threadIdx.x; f < 16 * (K / 4); f += 128) {
            int r = f / (K / 4);
            int q = f % (K / 4);
            float4 v = *(const float4*)(A + (size_t)(m0 + r) * K + q * 4);
            *(float4*)&As[r * LDA + q * 4] = v;
        }
        // ---- stage W tile transposed: WT[c][k] = W[k][cb0+c]
        for (int f = threadIdx.x; f < K * 16; f += 128) {
            int k = f >> 4;                // 16 float4 per k-row (64 cols)
            int q = f & 15;
            float4 v = *(const float4*)(W + (size_t)k * N + cb0 + q * 4);
            WTs[(q * 4 + 0) * LDA + k] = v.x;
            WTs[(q * 4 + 1) * LDA + k] = v.y;
            WTs[(q * 4 + 2) * LDA + k] = v.z;
            WTs[(q * 4 + 3) * LDA + k] = v.w;
        }
        __syncthreads();

        const float s = scale ? scale[row] : 1.0f;
        const float* aP = &As[lm * LDA + 2 * half];
        const float* bP = &WTs[(wave * 16 + lm) * LDA + 2 * half];

#pragma unroll 8
        for (int k = 0; k < K; k += 4) {
            v2f a = *(const v2f*)(aP + k);
            a.x *= s;
            a.y *= s;
            v2f b = *(const v2f*)(bP + k);
            acc = __builtin_amdgcn_wmma_f32_16x16x4_f32(
                /*neg_a=*/false, a, /*neg_b=*/false, b,
                /*c_mod=*/(short)0, acc, /*reuse_a=*/false, /*reuse_b=*/false);
        }
    }

    float bsum = bias0 ? bias0[col] : 0.0f;
    if (bias1) bsum += bias1[col];

#pragma unroll
    for (int v = 0; v < 8; ++v) {
        float val = acc[v] + bsum;                    // D: m = v + 8*half
        if (relu) val = fmaxf(val, 0.0f);
        out[(size_t)(m0 + v + 8 * half) * N + col] = val;
    }
}

// ---------------------------------------------------------------------------
extern "C" void kernel_launch(void* const* d_in, const int* in_sizes, int n_in,
                              void* d_out, int out_size, void* d_ws, size_t ws_size,
                              hipStream_t stream)
{
    (void)in_sizes; (void)n_in; (void)out_size; (void)ws_size;

    const float* x_robot = (const float*)d_in[0];
    const float* x_ball  = (const float*)d_in[1];
    const int*   ei_rr   = (const int*)d_in[2];
    const int*   ei_rb   = (const int*)d_in[3];
    const int*   ei_br   = (const int*)d_in[4];
    const float* Wl0_rr = (const float*)d_in[5];
    const float* Wr0_rr = (const float*)d_in[6];
    const float* b0_rr  = (const float*)d_in[7];
    const float* Wl0_rb = (const float*)d_in[8];
    const float* Wr0_rb = (const float*)d_in[9];
    const float* b0_rb  = (const float*)d_in[10];
    const float* Wl0_br = (const float*)d_in[11];
    const float* Wr0_br = (const float*)d_in[12];
    const float* b0_br  = (const float*)d_in[13];
    const float* Wl1_rr = (const float*)d_in[14];
    const float* Wr1_rr = (const float*)d_in[15];
    const float* b1_rr  = (const float*)d_in[16];
    // d_in[17..19] = rb layer-1 weights: unused (reference returns robot nodes)
    const float* Wl1_br = (const float*)d_in[20];
    const float* Wr1_br = (const float*)d_in[21];
    const float* b1_br  = (const float*)d_in[22];

    // Workspace: 4 node-feature buffers (100k x 128 f32 each) + 3 degree arrays.
    float* ws = (float*)d_ws;
    const size_t nodeF = (size_t)NR * DH;
    float* buf0 = ws;                 // agg_rb -> agg_rr -> agg1_rr
    float* buf1 = ws + 1 * nodeF;     // bl (ball hidden)
    float* buf2 = ws + 2 * nodeF;     // agg_br -> agg1_br
    float* buf3 = ws + 3 * nodeF;     // r (robot hidden)
    float* deg_rr = ws + 4 * nodeF;
    float* deg_br = deg_rr + NR;
    float* deg_rb = deg_br + NR;

    const int scatBlocks = (NE * 32 + 255) / 256;   // one wave per edge
    const size_t nodeBytes = nodeF * sizeof(float);

    // ---- stage A: ball aggregation (robot -> ball over ei_rb) ----
    hipMemsetAsync(buf0, 0, nodeBytes, stream);
    hipMemsetAsync(deg_rb, 0, NB * sizeof(float), stream);
    scatter_accum<<<scatBlocks, 256, 0, stream>>>(x_robot, ei_rb, ei_rb + NE, buf0, deg_rb, NE);
    invdeg_kernel<<<(NB + 255) / 256, 256, 0, stream>>>(deg_rb, NB);
    // bl = relu(mean_agg @ Wl0_rb + x_ball @ Wr0_rb + b0_rb)
    sage_wmma_gemm<DIN><<<dim3(NB / 16, DH / 64), 128, 0, stream>>>(
        buf0, Wl0_rb, deg_rb, x_ball, Wr0_rb,
        nullptr, nullptr, nullptr, nullptr, nullptr,
        b0_rb, nullptr, buf1, DH, 1);

    // ---- stage B: robot aggregations (rr + br) ----
    hipMemsetAsync(buf0, 0, nodeBytes, stream);
    hipMemsetAsync(buf2, 0, nodeBytes, stream);
    hipMemsetAsync(deg_rr, 0, NR * sizeof(float), stream);
    hipMemsetAsync(deg_br, 0, NR * sizeof(float), stream);
    scatter_accum<<<scatBlocks, 256, 0, stream>>>(x_robot, ei_rr, ei_rr + NE, buf0, deg_rr, NE);
    scatter_accum<<<scatBlocks, 256, 0, stream>>>(x_ball,  ei_br, ei_br + NE, buf2, deg_br, NE);
    invdeg_kernel<<<(NR + 255) / 256, 256, 0, stream>>>(deg_rr, NR);
    invdeg_kernel<<<(NR + 255) / 256, 256, 0, stream>>>(deg_br, NR);
    // r = relu(agg_rr@Wl0_rr + x_robot@Wr0_rr + agg_br@Wl0_br + x_robot@Wr0_br + b0_rr + b0_br)
    sage_wmma_gemm<DIN><<<dim3(NR / 16, DH / 64), 128, 0, stream>>>(
        buf0, Wl0_rr, deg_rr, x_robot, Wr0_rr,
        buf2, Wl0_br, deg_br, x_robot, Wr0_br,
        b0_rr, b0_br, buf3, DH, 1);

    // ---- stage C: layer-1 aggregations (same graphs -> reuse inverse degrees) ----
    hipMemsetAsync(buf0, 0, nodeBytes, stream);
    hipMemsetAsync(buf2, 0, nodeBytes, stream);
    scatter_accum<<<scatBlocks, 256, 0, stream>>>(buf3, ei_rr, ei_rr + NE, buf0, nullptr, NE);
    scatter_accum<<<scatBlocks, 256, 0, stream>>>(buf1, ei_br, ei_br + NE, buf2, nullptr, NE);
    // out = agg1_rr@Wl1_rr + r@Wr1_rr + agg1_br@Wl1_br + r@Wr1_br + b1_rr + b1_br
    sage_wmma_gemm<DH><<<dim3(NR / 16, DOUT / 64), 128, 0, stream>>>(
        buf0, Wl1_rr, deg_rr, buf3, Wr1_rr,
        buf2, Wl1_br, deg_br, buf3, Wr1_br,
        b1_rr, b1_br, (float*)d_out, DOUT, 0);
}